// DynamicGraphCNN_61194694033708
// MI455X (gfx1250) — compile-verified
//
#include <hip/hip_runtime.h>

typedef __attribute__((ext_vector_type(2))) float v2f;
typedef __attribute__((ext_vector_type(8))) float v8f;

#define BATCH 8
#define NPTS  2048
#define KNN_K 20
#define EPSI  1e-5f
#define SLOPE 0.2f

// number of y-groups per batch for the stats kernels (each group = 8 n-tiles)
#define NGRP   16
#define NTILES (BATCH * NGRP)  // 128 partial slots per channel

// FP32 WMMA: D(16x16) = A(16x4) * B(4x16) + C  -> v_wmma_f32_16x16x4_f32
__device__ __forceinline__ v8f wmma4(v2f a, v2f b, v8f c) {
  return __builtin_amdgcn_wmma_f32_16x16x4_f32(false, a, false, b, (short)0, c,
                                               false, false);
}

// ---------------------------------------------------------------------------
// prep kernels
// ---------------------------------------------------------------------------
__global__ void transpose_xyz_kernel(const float* __restrict__ xyz,
                                     float* __restrict__ x0) {
  int i = blockIdx.x * blockDim.x + threadIdx.x;  // b*N + n
  if (i >= BATCH * NPTS) return;
  int b = i / NPTS, n = i - b * NPTS;
#pragma unroll
  for (int c = 0; c < 4; ++c)
    x0[((size_t)b * 4 + c) * NPTS + n] = (c < 3) ? xyz[(size_t)i * 3 + c] : 0.f;
}

__global__ void zero_xfpad_kernel(float* __restrict__ xf) {
  int i = blockIdx.x * blockDim.x + threadIdx.x;  // over B * 3 * N
  if (i >= BATCH * 3 * NPTS) return;
  int b = i / (3 * NPTS), r = i - b * 3 * NPTS;
  xf[(size_t)b * 516 * NPTS + (size_t)513 * NPTS + r] = 0.f;
}

__global__ void pad_weight_kernel(const float* __restrict__ W,
                                  float* __restrict__ Wp, int O, int C,
                                  int Opad, int Cpad) {
  int i = blockIdx.x * blockDim.x + threadIdx.x;
  if (i >= Opad * Cpad) return;
  int r = i / Cpad, c = i - r * Cpad;
  Wp[i] = (r < O && c < C) ? W[(size_t)r * C + c] : 0.f;
}

__global__ void sqnorm_kernel(const float* __restrict__ x,
                              float* __restrict__ nrm, int Cin, size_t xBS) {
  int i = blockIdx.x * blockDim.x + threadIdx.x;
  if (i >= BATCH * NPTS) return;
  int b = i / NPTS, n = i - b * NPTS;
  const float* xb = x + (size_t)b * xBS;
  float s = 0.f;
  for (int c = 0; c < Cin; ++c) {
    float t = xb[(size_t)c * NPTS + n];
    s += t * t;
  }
  nrm[i] = s;
}

// ---------------------------------------------------------------------------
// KNN: one wave per 16 query points. Query tile staged in LDS (reused across
// all 128 key tiles); Gram via FP32 WMMA; stable register-resident top-20.
// Feature arrays are zero-padded so all loads are unconditional.
// ---------------------------------------------------------------------------
template <int CPAD>
__global__ __launch_bounds__(32) void knn_topk_kernel(
    const float* __restrict__ x, const float* __restrict__ nrm,
    int* __restrict__ knnOut, size_t xBS) {
  constexpr int LSTR = CPAD + 2;
  __shared__ float lq[16 * LSTR];
  __shared__ float pd[16][17];
  const int lane = threadIdx.x, col = lane & 15, half = lane >> 4;
  const int b = blockIdx.y, q0 = blockIdx.x * 16;
  const float* xb = x + (size_t)b * xBS;
  const float* nb = nrm + (size_t)b * NPTS;

  for (int i = lane; i < 16 * CPAD; i += 32) {
    int r = i / CPAD, c = i - r * CPAD;
    lq[r * LSTR + c] = xb[(size_t)c * NPTS + q0 + r];
  }
  __syncthreads();

  float topv[KNN_K];
  int topi[KNN_K];
#pragma unroll
  for (int i = 0; i < KNN_K; ++i) { topv[i] = -3.4e38f; topi[i] = 0; }

  float qn[8];
#pragma unroll
  for (int v = 0; v < 8; ++v) qn[v] = nb[q0 + v + 8 * half];

  for (int kb0 = 0; kb0 < NPTS; kb0 += 16) {
    v8f acc = {};
#pragma unroll
    for (int c0 = 0; c0 < CPAD; c0 += 4) {
      const int ck = c0 + half * 2;
      v2f av = *(const v2f*)(lq + col * LSTR + ck);
      v2f bv;
      bv.x = xb[(size_t)ck * NPTS + kb0 + col];
      bv.y = xb[(size_t)(ck + 1) * NPTS + kb0 + col];
      acc = wmma4(av, bv, acc);
    }
    float kn = nb[kb0 + col];
#pragma unroll
    for (int v = 0; v < 8; ++v)
      pd[v + 8 * half][col] = 2.f * acc[v] - qn[v] - kn;
    __syncthreads();
    if (lane < 16) {
      for (int t = 0; t < 16; ++t) {
        float vv = pd[lane][t];
        if (vv > topv[KNN_K - 1]) {
          topv[KNN_K - 1] = vv;
          topi[KNN_K - 1] = kb0 + t;
#pragma unroll
          for (int j = KNN_K - 1; j > 0; --j) {
            if (topv[j] > topv[j - 1]) {
              float tv = topv[j]; topv[j] = topv[j - 1]; topv[j - 1] = tv;
              int ti = topi[j]; topi[j] = topi[j - 1]; topi[j - 1] = ti;
            }
          }
        }
      }
    }
    __syncthreads();
  }
  if (lane < 16) {
    size_t base = ((size_t)(b * NPTS + q0 + lane)) * KNN_K;
#pragma unroll
    for (int i = 0; i < KNN_K; ++i) knnOut[base + i] = topi[i];
  }
}

// ---------------------------------------------------------------------------
// branchless edge feature: always-valid loads + value selects (v_cndmask)
// c < CIN      -> x[c][j] - x[c][n]
// c < 2*CIN    -> x[c-CIN][n]
// else         -> 0 (pad)
// ---------------------------------------------------------------------------
template <int CIN>
__device__ __forceinline__ float edge_val(const float* __restrict__ xb, int c,
                                          int n, int j) {
  const bool isDiff = (c < CIN);
  const bool isCtr = (c < 2 * CIN);
  int cj = isDiff ? c : 0;
  int cn = isDiff ? c : (c - CIN);
  cn = (cn < CIN) ? cn : 0;
  float fj = xb[(size_t)cj * NPTS + j];
  float fn = xb[(size_t)cn * NPTS + n];
  return isDiff ? (fj - fn) : (isCtr ? fn : 0.f);
}

// A-operand (weights) from LDS, B-operand gathered from features; unrolled.
template <int CIN, int CEPAD>
__device__ __forceinline__ v8f edge_tile_lds(const float* __restrict__ lw,
                                             const float* __restrict__ xb,
                                             int n, int j, int col, int half) {
  constexpr int LSTR = CEPAD + 2;
  v8f acc = {};
#pragma unroll
  for (int c0 = 0; c0 < CEPAD; c0 += 4) {
    const int ck = c0 + half * 2;
    v2f av = *(const v2f*)(lw + col * LSTR + ck);
    v2f bv;
    bv.x = edge_val<CIN>(xb, ck, n, j);
    bv.y = edge_val<CIN>(xb, ck + 1, n, j);
    acc = wmma4(av, bv, acc);
  }
  return acc;
}

template <int CEPAD>
__device__ __forceinline__ void stage_w(const float* __restrict__ Wp, int o0,
                                        float* __restrict__ lw) {
  constexpr int LSTR = CEPAD + 2;
  for (int i = threadIdx.x; i < 16 * CEPAD; i += 32) {
    int r = i / CEPAD, c = i - r * CEPAD;
    lw[r * LSTR + c] = Wp[(size_t)(o0 + r) * CEPAD + c];
  }
  __syncthreads();
}

// pass 1: deterministic sum / sumsq partials (one group = 8 n-tiles)
template <int CIN, int CEPAD>
__global__ __launch_bounds__(32) void conv2d_stats_kernel(
    const float* __restrict__ Wp, const float* __restrict__ x,
    const int* __restrict__ knn, float* __restrict__ ps1,
    float* __restrict__ ps2, int O, size_t xBS) {
  constexpr int LSTR = CEPAD + 2;
  __shared__ float lw[16 * LSTR];
  __shared__ float l1[16][16], l2[16][16];
  const int lane = threadIdx.x, col = lane & 15, half = lane >> 4;
  const int o0 = blockIdx.x * 16;
  const int gy = blockIdx.y;
  const int b = gy / NGRP, grp = gy - b * NGRP;
  const float* xb = x + (size_t)b * xBS;

  stage_w<CEPAD>(Wp, o0, lw);

  float s1[8] = {0, 0, 0, 0, 0, 0, 0, 0};
  float s2[8] = {0, 0, 0, 0, 0, 0, 0, 0};
  for (int t = 0; t < NPTS / 16 / NGRP; ++t) {
    const int n = (grp * (NPTS / 16 / NGRP) + t) * 16 + col;
    const int* kp = knn + ((size_t)(b * NPTS + n)) * KNN_K;
    for (int k = 0; k < KNN_K; ++k) {
      int j = kp[k];
      v8f acc = edge_tile_lds<CIN, CEPAD>(lw, xb, n, j, col, half);
#pragma unroll
      for (int v = 0; v < 8; ++v) {
        float y = acc[v];
        s1[v] += y;
        s2[v] += y * y;
      }
    }
  }
#pragma unroll
  for (int v = 0; v < 8; ++v) {
    l1[v + 8 * half][col] = s1[v];
    l2[v + 8 * half][col] = s2[v];
  }
  __syncthreads();
  if (lane < 16) {
    int o = o0 + lane;
    if (o < O) {
      float a = 0.f, q = 0.f;
#pragma unroll
      for (int t = 0; t < 16; ++t) { a += l1[lane][t]; q += l2[lane][t]; }
      ps1[(size_t)o * NTILES + gy] = a;
      ps2[(size_t)o * NTILES + gy] = q;
    }
  }
}

__global__ void bn_finalize_kernel(const float* __restrict__ ps1,
                                   const float* __restrict__ ps2,
                                   const float* __restrict__ g,
                                   const float* __restrict__ bb,
                                   float* __restrict__ bnA,
                                   float* __restrict__ bnC, int O, float invM) {
  int o = blockIdx.x * blockDim.x + threadIdx.x;
  if (o >= O) return;
  float s = 0.f, q = 0.f;
  for (int t = 0; t < NTILES; ++t) {
    s += ps1[(size_t)o * NTILES + t];
    q += ps2[(size_t)o * NTILES + t];
  }
  float m = s * invM;
  float var = q * invM - m * m;
  float a = g[o] * rsqrtf(var + EPSI);
  bnA[o] = a;
  bnC[o] = bb[o] - m * a;
}

// pass 2: GEMM + BN + leaky-relu + max over K neighbors
template <int CIN, int CEPAD>
__global__ __launch_bounds__(32) void conv2d_apply_kernel(
    const float* __restrict__ Wp, const float* __restrict__ x,
    const int* __restrict__ knn, const float* __restrict__ bnA,
    const float* __restrict__ bnC, float* __restrict__ out, size_t outBS, int O,
    size_t xBS) {
  constexpr int LSTR = CEPAD + 2;
  __shared__ float lw[16 * LSTR];
  const int lane = threadIdx.x, col = lane & 15, half = lane >> 4;
  const int o0 = blockIdx.x * 16;
  const int gy = blockIdx.y;
  const int b = gy / NGRP, grp = gy - b * NGRP;
  const float* xb = x + (size_t)b * xBS;

  stage_w<CEPAD>(Wp, o0, lw);

  float aco[8], cco[8];
#pragma unroll
  for (int v = 0; v < 8; ++v) {
    int o = o0 + v + 8 * half;  // < 1024 alloc, safe unguarded
    aco[v] = bnA[o];
    cco[v] = bnC[o];
  }
  float* ob = out + (size_t)b * outBS;
  for (int t = 0; t < NPTS / 16 / NGRP; ++t) {
    const int n = (grp * (NPTS / 16 / NGRP) + t) * 16 + col;
    const int* kp = knn + ((size_t)(b * NPTS + n)) * KNN_K;
    float mx[8];
#pragma unroll
    for (int v = 0; v < 8; ++v) mx[v] = -3.4e38f;
    for (int k = 0; k < KNN_K; ++k) {
      int j = kp[k];
      v8f acc = edge_tile_lds<CIN, CEPAD>(lw, xb, n, j, col, half);
#pragma unroll
      for (int v = 0; v < 8; ++v) {
        float z = acc[v] * aco[v] + cco[v];
        z = (z > 0.f) ? z : SLOPE * z;
        mx[v] = fmaxf(mx[v], z);
      }
    }
#pragma unroll
    for (int v = 0; v < 8; ++v) {
      int o = o0 + v + 8 * half;
      if (o < O) ob[(size_t)o * NPTS + n] = mx[v];
    }
  }
}

// ---------------------------------------------------------------------------
// layer 5 (1D conv 513->1024): dense GEMM, W tile in LDS
// ---------------------------------------------------------------------------
#define CFPAD 516
#define LSTR5 (CFPAD + 2)

__device__ __forceinline__ v8f feat_tile_lds(const float* __restrict__ lw,
                                             const float* __restrict__ xb,
                                             int n, int col, int half) {
  v8f acc = {};
#pragma unroll
  for (int c0 = 0; c0 < CFPAD; c0 += 4) {
    const int ck = c0 + half * 2;
    v2f av = *(const v2f*)(lw + col * LSTR5 + ck);
    v2f bv;
    bv.x = xb[(size_t)ck * NPTS + n];
    bv.y = xb[(size_t)(ck + 1) * NPTS + n];
    acc = wmma4(av, bv, acc);
  }
  return acc;
}

__global__ __launch_bounds__(32) void conv1d_stats_kernel(
    const float* __restrict__ Wp, const float* __restrict__ x,
    float* __restrict__ ps1, float* __restrict__ ps2, size_t xBS) {
  __shared__ float lw[16 * LSTR5];
  __shared__ float l1[16][16], l2[16][16];
  const int lane = threadIdx.x, col = lane & 15, half = lane >> 4;
  const int o0 = blockIdx.x * 16;
  const int gy = blockIdx.y;
  const int b = gy / NGRP, grp = gy - b * NGRP;
  const float* xb = x + (size_t)b * xBS;

  for (int i = lane; i < 16 * CFPAD; i += 32) {
    int r = i / CFPAD, c = i - r * CFPAD;
    lw[r * LSTR5 + c] = Wp[(size_t)(o0 + r) * CFPAD + c];
  }
  __syncthreads();

  float s1[8] = {0, 0, 0, 0, 0, 0, 0, 0};
  float s2[8] = {0, 0, 0, 0, 0, 0, 0, 0};
  for (int t = 0; t < NPTS / 16 / NGRP; ++t) {
    const int n = (grp * (NPTS / 16 / NGRP) + t) * 16 + col;
    v8f acc = feat_tile_lds(lw, xb, n, col, half);
#pragma unroll
    for (int v = 0; v < 8; ++v) {
      float y = acc[v];
      s1[v] += y;
      s2[v] += y * y;
    }
  }
#pragma unroll
  for (int v = 0; v < 8; ++v) {
    l1[v + 8 * half][col] = s1[v];
    l2[v + 8 * half][col] = s2[v];
  }
  __syncthreads();
  if (lane < 16) {
    int o = o0 + lane;
    float a = 0.f, q = 0.f;
#pragma unroll
    for (int t = 0; t < 16; ++t) { a += l1[lane][t]; q += l2[lane][t]; }
    ps1[(size_t)o * NTILES + gy] = a;
    ps2[(size_t)o * NTILES + gy] = q;
  }
}

__global__ __launch_bounds__(32) void conv1d_apply_kernel(
    const float* __restrict__ Wp, const float* __restrict__ x,
    const float* __restrict__ bnA, const float* __restrict__ bnC,
    float* __restrict__ gmx, float* __restrict__ gav, size_t xBS) {
  __shared__ float lw[16 * LSTR5];
  __shared__ float lm[16][16], ls[16][16];
  const int lane = threadIdx.x, col = lane & 15, half = lane >> 4;
  const int o0 = blockIdx.x * 16;
  const int b = blockIdx.y;
  const float* xb = x + (size_t)b * xBS;

  for (int i = lane; i < 16 * CFPAD; i += 32) {
    int r = i / CFPAD, c = i - r * CFPAD;
    lw[r * LSTR5 + c] = Wp[(size_t)(o0 + r) * CFPAD + c];
  }
  __syncthreads();

  float aco[8], cco[8], mx[8], sm[8];
#pragma unroll
  for (int v = 0; v < 8; ++v) {
    int o = o0 + v + 8 * half;
    aco[v] = bnA[o];
    cco[v] = bnC[o];
    mx[v] = -3.4e38f;
    sm[v] = 0.f;
  }
  for (int nt = 0; nt < NPTS / 16; ++nt) {
    int n = nt * 16 + col;
    v8f acc = feat_tile_lds(lw, xb, n, col, half);
#pragma unroll
    for (int v = 0; v < 8; ++v) {
      float z = acc[v] * aco[v] + cco[v];
      z = (z > 0.f) ? z : SLOPE * z;
      mx[v] = fmaxf(mx[v], z);
      sm[v] += z;
    }
  }
#pragma unroll
  for (int v = 0; v < 8; ++v) {
    lm[v + 8 * half][col] = mx[v];
    ls[v + 8 * half][col] = sm[v];
  }
  __syncthreads();
  if (lane < 16) {
    int o = o0 + lane;
    float m = -3.4e38f, s = 0.f;
#pragma unroll
    for (int t = 0; t < 16; ++t) {
      m = fmaxf(m, lm[lane][t]);
      s += ls[lane][t];
    }
    gmx[(size_t)b * 1024 + o] = m;
    gav[(size_t)b * 1024 + o] = s * (1.f / (float)NPTS);
  }
}

// ---------------------------------------------------------------------------
// final concat: [gmax(1024), gavg(1024), x0(3), x_f(513)] x N
// ---------------------------------------------------------------------------
__global__ void assemble_out_kernel(const float* __restrict__ gmx,
                                    const float* __restrict__ gav,
                                    const float* __restrict__ x0,
                                    const float* __restrict__ xf,
                                    float* __restrict__ out, size_t total) {
  size_t i = (size_t)blockIdx.x * blockDim.x + threadIdx.x;
  if (i >= total) return;
  int n = (int)(i % NPTS);
  size_t r = i / NPTS;
  int ch = (int)(r % 2564);
  int b = (int)(r / 2564);
  float v;
  if (ch < 1024)
    v = gmx[(size_t)b * 1024 + ch];
  else if (ch < 2048)
    v = gav[(size_t)b * 1024 + (ch - 1024)];
  else if (ch < 2051)
    v = x0[((size_t)b * 4 + (ch - 2048)) * NPTS + n];
  else
    v = xf[((size_t)b * 516 + (ch - 2051)) * NPTS + n];
  out[i] = v;
}

// ---------------------------------------------------------------------------
extern "C" void kernel_launch(void* const* d_in, const int* in_sizes, int n_in,
                              void* d_out, int out_size, void* d_ws,
                              size_t ws_size, hipStream_t stream) {
  const float* xyz = (const float*)d_in[0];
  const float* W1 = (const float*)d_in[1];
  const float* g1 = (const float*)d_in[2];
  const float* b1 = (const float*)d_in[3];
  const float* W2 = (const float*)d_in[4];
  const float* g2 = (const float*)d_in[5];
  const float* b2 = (const float*)d_in[6];
  const float* W3 = (const float*)d_in[7];
  const float* g3 = (const float*)d_in[8];
  const float* b3 = (const float*)d_in[9];
  const float* W4 = (const float*)d_in[10];
  const float* g4 = (const float*)d_in[11];
  const float* b4 = (const float*)d_in[12];
  const float* W5 = (const float*)d_in[13];
  const float* g5 = (const float*)d_in[14];
  const float* b5 = (const float*)d_in[15];
  float* out = (float*)d_out;

  // workspace partition (floats)
  float* p = (float*)d_ws;
  float* x0 = p;   p += (size_t)BATCH * 4 * NPTS;           // padded to 4 ch
  float* xf = p;   p += (size_t)BATCH * CFPAD * NPTS;       // 516 ch (513+pad)
  float* nrm = p;  p += (size_t)BATCH * NPTS;
  float* bnA = p;  p += 1024;
  float* bnC = p;  p += 1024;
  float* gmx = p;  p += (size_t)BATCH * 1024;
  float* gav = p;  p += (size_t)BATCH * 1024;
  float* ps1 = p;  p += (size_t)1024 * NTILES;
  float* ps2 = p;  p += (size_t)1024 * NTILES;
  float* wbuf = p; p += (size_t)1024 * CFPAD;               // padded weights
  int* knn = (int*)p;

  const size_t x0BS = (size_t)4 * NPTS;
  const size_t xfBS = (size_t)CFPAD * NPTS;
  const dim3 knnGrid(NPTS / 16, BATCH);
  const dim3 statGrid2(0, 0);
  const float invM2d = 1.f / (float)((size_t)BATCH * NPTS * KNN_K);
  const float invM1d = 1.f / (float)((size_t)BATCH * NPTS);
  const int thr = 256;

  transpose_xyz_kernel<<<(BATCH * NPTS + thr - 1) / thr, thr, 0, stream>>>(xyz, x0);
  zero_xfpad_kernel<<<(BATCH * 3 * NPTS + thr - 1) / thr, thr, 0, stream>>>(xf);

  // ---- layer 1: Cin=3 (edge 6 -> pad 8), O=64, in x0, out xf[0:64]
  pad_weight_kernel<<<(64 * 8 + thr - 1) / thr, thr, 0, stream>>>(W1, wbuf, 64, 6, 64, 8);
  sqnorm_kernel<<<(BATCH * NPTS + thr - 1) / thr, thr, 0, stream>>>(x0, nrm, 3, x0BS);
  knn_topk_kernel<4><<<knnGrid, 32, 0, stream>>>(x0, nrm, knn, x0BS);
  conv2d_stats_kernel<3, 8><<<dim3(4, NTILES), 32, 0, stream>>>(wbuf, x0, knn, ps1, ps2, 64, x0BS);
  bn_finalize_kernel<<<1, 64, 0, stream>>>(ps1, ps2, g1, b1, bnA, bnC, 64, invM2d);
  conv2d_apply_kernel<3, 8><<<dim3(4, NTILES), 32, 0, stream>>>(
      wbuf, x0, knn, bnA, bnC, xf, xfBS, 64, x0BS);

  // ---- layer 2: Cin=64 (edge 128), O=64, in xf[0:64], out xf[64:128]
  pad_weight_kernel<<<(64 * 128 + thr - 1) / thr, thr, 0, stream>>>(W2, wbuf, 64, 128, 64, 128);
  sqnorm_kernel<<<(BATCH * NPTS + thr - 1) / thr, thr, 0, stream>>>(xf, nrm, 64, xfBS);
  knn_topk_kernel<64><<<knnGrid, 32, 0, stream>>>(xf, nrm, knn, xfBS);
  conv2d_stats_kernel<64, 128><<<dim3(4, NTILES), 32, 0, stream>>>(wbuf, xf, knn, ps1, ps2, 64, xfBS);
  bn_finalize_kernel<<<1, 64, 0, stream>>>(ps1, ps2, g2, b2, bnA, bnC, 64, invM2d);
  conv2d_apply_kernel<64, 128><<<dim3(4, NTILES), 32, 0, stream>>>(
      wbuf, xf, knn, bnA, bnC, xf + (size_t)64 * NPTS, xfBS, 64, xfBS);

  // ---- layer 3: Cin=64 (edge 128), O=128, in xf[64:128], out xf[128:256]
  pad_weight_kernel<<<(128 * 128 + thr - 1) / thr, thr, 0, stream>>>(W3, wbuf, 128, 128, 128, 128);
  sqnorm_kernel<<<(BATCH * NPTS + thr - 1) / thr, thr, 0, stream>>>(
      xf + (size_t)64 * NPTS, nrm, 64, xfBS);
  knn_topk_kernel<64><<<knnGrid, 32, 0, stream>>>(xf + (size_t)64 * NPTS, nrm, knn, xfBS);
  conv2d_stats_kernel<64, 128><<<dim3(8, NTILES), 32, 0, stream>>>(
      wbuf, xf + (size_t)64 * NPTS, knn, ps1, ps2, 128, xfBS);
  bn_finalize_kernel<<<1, 128, 0, stream>>>(ps1, ps2, g3, b3, bnA, bnC, 128, invM2d);
  conv2d_apply_kernel<64, 128><<<dim3(8, NTILES), 32, 0, stream>>>(
      wbuf, xf + (size_t)64 * NPTS, knn, bnA, bnC, xf + (size_t)128 * NPTS, xfBS, 128, xfBS);

  // ---- layer 4: Cin=128 (edge 256), O=257 (pad 272), in xf[128:256], out xf[256:513]
  pad_weight_kernel<<<(272 * 256 + thr - 1) / thr, thr, 0, stream>>>(W4, wbuf, 257, 256, 272, 256);
  sqnorm_kernel<<<(BATCH * NPTS + thr - 1) / thr, thr, 0, stream>>>(
      xf + (size_t)128 * NPTS, nrm, 128, xfBS);
  knn_topk_kernel<128><<<knnGrid, 32, 0, stream>>>(xf + (size_t)128 * NPTS, nrm, knn, xfBS);
  conv2d_stats_kernel<128, 256><<<dim3(17, NTILES), 32, 0, stream>>>(
      wbuf, xf + (size_t)128 * NPTS, knn, ps1, ps2, 257, xfBS);
  bn_finalize_kernel<<<(257 + 63) / 64, 64, 0, stream>>>(ps1, ps2, g4, b4, bnA, bnC, 257, invM2d);
  conv2d_apply_kernel<128, 256><<<dim3(17, NTILES), 32, 0, stream>>>(
      wbuf, xf + (size_t)128 * NPTS, knn, bnA, bnC, xf + (size_t)256 * NPTS, xfBS, 257, xfBS);

  // ---- layer 5: 1D conv 513 -> 1024 (pad 516) + global max/avg
  pad_weight_kernel<<<(1024 * CFPAD + thr - 1) / thr, thr, 0, stream>>>(
      W5, wbuf, 1024, 513, 1024, CFPAD);
  conv1d_stats_kernel<<<dim3(64, NTILES), 32, 0, stream>>>(wbuf, xf, ps1, ps2, xfBS);
  bn_finalize_kernel<<<16, 64, 0, stream>>>(ps1, ps2, g5, b5, bnA, bnC, 1024, invM1d);
  conv1d_apply_kernel<<<dim3(64, BATCH), 32, 0, stream>>>(wbuf, xf, bnA, bnC, gmx, gav, xfBS);

  // ---- output assembly (B, 2564, N)
  size_t total = (size_t)BATCH * 2564 * NPTS;
  assemble_out_kernel<<<(unsigned)((total + thr - 1) / thr), thr, 0, stream>>>(
      gmx, gav, x0, xf, out, total);
}